// MinCostMatcher_52261162058614
// MI455X (gfx1250) — compile-verified
//
#include <hip/hip_runtime.h>
#include <hip/hip_bf16.h>

// ---------------------------------------------------------------------------
// MinCostMatcher for MI455X (gfx1250, wave32).
//   C[b,q,t] = 5*L1(norm boxes) + 2*focal_class_cost - 2*rotated_IoU
//   src_ind[b,t] = argmin_q C ; tgt_ind[b,t] = t
// Class cost is a one-hot contraction F[hw,80] @ O[80,24] done with
// V_WMMA_F32_16X16X4_F32. Rotated IoU is a Sutherland-Hodgman convex clip
// (the VALU-bound hot loop). Hot-path divides -> v_rcp_f32; logs -> bare
// v_log_f32 (log2) with ln2 folded into the focal constants.
// ---------------------------------------------------------------------------

typedef __attribute__((ext_vector_type(2))) float v2f;
typedef __attribute__((ext_vector_type(8))) float v8f;

#define BS   4
#define KCLS 80
#define HW   (112*112)       // 12544 = 49*256 = 784*16
#define TT   24
#define QT_PER_B (HW/16)     // 784

__device__ __forceinline__ float fast_rcp(float x) {
  return __builtin_amdgcn_rcpf(x);   // v_rcp_f32
}
__device__ __forceinline__ float fast_log2(float x) {
  return __builtin_amdgcn_logf(x);   // bare v_log_f32 (= log2), args never subnormal
}

__device__ __forceinline__ float focal2(float l) {
  // 2*(pos-neg) with W_CLASS, ALPHA and ln2 folded:
  //   = 1.5*ln2 * p^2 * log2(1-p+eps) - 0.5*ln2 * (1-p)^2 * log2(p+eps)
  float p   = fast_rcp(1.f + __expf(-l));
  float om  = 1.f - p;
  float l2p = fast_log2(p  + 1e-8f);
  float l2o = fast_log2(om + 1e-8f);
  return 1.03972077084f * p * p * l2o - 0.34657359028f * om * om * l2p;
}

// ---------------------------------------------------------------------------
// Kernel 0: per-target precompute (corners, area, normalized box) + argmin init
// ws layout: [0,768) u64 gmin slots (96); [1024, 1024+6144) target data 16f each
// ---------------------------------------------------------------------------
__global__ void prep_kernel(const float* __restrict__ tboxes,
                            const float* __restrict__ imgt,
                            float* __restrict__ wsT,
                            unsigned long long* __restrict__ gmin) {
  int i = blockIdx.x * blockDim.x + threadIdx.x;
  if (i >= BS * TT) return;
  gmin[i] = ~0ull;
  const float* bx = tboxes + (size_t)i * 5;
  float cx = bx[0], cy = bx[1], w = bx[2], h = bx[3];
  float th = bx[4] * 0.017453292519943295f;
  float c = cosf(th), s = sinf(th);
  float hw2 = 0.5f * w, hh2 = 0.5f * h;
  float* o = wsT + (size_t)i * 16;
  // cv2 RotatedRect corner order (dx = {+,+,-,-}, dy = {+,-,-,+})
  o[0] = cx + c*hw2 - s*hh2;  o[1] = cy + s*hw2 + c*hh2;
  o[2] = cx + c*hw2 + s*hh2;  o[3] = cy + s*hw2 - c*hh2;
  o[4] = cx - c*hw2 + s*hh2;  o[5] = cy - s*hw2 - c*hh2;
  o[6] = cx - c*hw2 - s*hh2;  o[7] = cy - s*hw2 + c*hh2;
  o[8] = w * h;
  #pragma unroll
  for (int j = 0; j < 5; ++j) o[9 + j] = bx[j] / imgt[(size_t)i * 5 + j];
  o[14] = 0.f; o[15] = 0.f;
}

// ---------------------------------------------------------------------------
// Kernel 1: class cost via fp32 WMMA one-hot contraction.
// Each wave: 16 queries x 24 targets. A = focal costs (16x4 per step, built
// in-register from coalesced logit loads), B = one-hot of target ids.
// Writes C := 2*cost_class (rest of C added by main kernel).
// ---------------------------------------------------------------------------
__global__ void class_cost_kernel(const float* __restrict__ logits,
                                  const int* __restrict__ labels,
                                  float* __restrict__ Cout) {
  int wave  = (int)((blockIdx.x * blockDim.x + threadIdx.x) >> 5);
  int lane  = (int)(threadIdx.x & 31);
  int b     = wave / QT_PER_B;
  int q0    = (wave % QT_PER_B) * 16;
  int mlane = lane & 15;      // M for A / N for B,D
  int hf    = lane >> 4;      // lane-half selects K pair {0,1} vs {2,3}
  int q     = q0 + mlane;

  const float* lb = logits + (size_t)b * KCLS * HW;
  int id0 = labels[b * TT + mlane] - 1;                               // targets 0..15
  int id1 = (mlane < 8) ? (labels[b * TT + 16 + mlane] - 1) : -1000;  // 16..23 (+pad)

  v8f acc0 = {}; v8f acc1 = {};
  #pragma unroll 4
  for (int kc = 0; kc < KCLS; kc += 4) {
    int c0 = kc + 2 * hf;
    // A fragment (16x4 f32): VGPR0 holds K={0|2}, VGPR1 holds K={1|3}
    float l0 = lb[(size_t)c0 * HW + q];
    float l1 = lb[(size_t)(c0 + 1) * HW + q];
    v2f a;  a.x = focal2(l0);           a.y = focal2(l1);
    // B fragments (4x16 f32 one-hot): row K striped same way, N = mlane
    v2f b0; b0.x = (id0 == c0) ? 1.f : 0.f;  b0.y = (id0 == c0 + 1) ? 1.f : 0.f;
    v2f b1; b1.x = (id1 == c0) ? 1.f : 0.f;  b1.y = (id1 == c0 + 1) ? 1.f : 0.f;
    acc0 = __builtin_amdgcn_wmma_f32_16x16x4_f32(false, a, false, b0, (short)0, acc0, false, false);
    acc1 = __builtin_amdgcn_wmma_f32_16x16x4_f32(false, a, false, b1, (short)0, acc1, false, false);
  }
  // D layout: VGPR v -> M = v + 8*half, N = mlane
  size_t base = ((size_t)b * HW + q0) * TT;
  #pragma unroll
  for (int v = 0; v < 8; ++v) {
    int M = v + 8 * hf;
    Cout[base + (size_t)M * TT + mlane] = acc0[v];
  }
  if (mlane < 8) {
    #pragma unroll
    for (int v = 0; v < 8; ++v) {
      int M = v + 8 * hf;
      Cout[base + (size_t)M * TT + 16 + mlane] = acc1[v];
    }
  }
}

// ---------------------------------------------------------------------------
// Convex quad intersection area: Sutherland-Hodgman clip of quad1 by quad2.
// Corner winding is fixed by construction -> inside test is d <= 0.
// ---------------------------------------------------------------------------
__device__ float quad_inter_area(const float* c1x, const float* c1y,
                                 const float* __restrict__ c2) {
  float px[12], py[12], qx[12], qy[12];
  int n = 4;
  #pragma unroll
  for (int i = 0; i < 4; ++i) { px[i] = c1x[i]; py[i] = c1y[i]; }
  for (int e = 0; e < 4; ++e) {
    float ax = c2[2 * e],            ay = c2[2 * e + 1];
    float bx = c2[(2 * e + 2) & 7],  by = c2[(2 * e + 3) & 7];
    float ex = bx - ax, ey = by - ay;
    int m = 0;
    float xp = px[n - 1], yp = py[n - 1];
    float dp = ex * (yp - ay) - ey * (xp - ax);
    for (int i = 0; i < n; ++i) {
      float xc = px[i], yc = py[i];
      float d = ex * (yc - ay) - ey * (xc - ax);
      bool inC = (d  <= 0.f);
      bool inP = (dp <= 0.f);
      if (inC != inP) {
        float den = dp - d;
        float tt  = (fabsf(den) > 1e-12f) ? (dp * fast_rcp(den)) : 0.f;
        qx[m] = xp + tt * (xc - xp);
        qy[m] = yp + tt * (yc - yp);
        ++m;
      }
      if (inC) { qx[m] = xc; qy[m] = yc; ++m; }
      dp = d; xp = xc; yp = yc;
    }
    n = m;
    if (n == 0) return 0.f;
    for (int i = 0; i < n; ++i) { px[i] = qx[i]; py[i] = qy[i]; }
  }
  float s = 0.f;
  for (int i = 0; i < n; ++i) {
    int j = (i + 1 == n) ? 0 : i + 1;
    s += px[i] * py[j] - py[i] * px[j];
  }
  return 0.5f * fabsf(s);
}

__device__ __forceinline__ unsigned order_f32(float f) {
  unsigned u = __float_as_uint(f);
  return (u & 0x80000000u) ? ~u : (u | 0x80000000u);  // monotone unsigned key
}

// ---------------------------------------------------------------------------
// Kernel 2: fused bbox-L1 + rotated IoU + total cost + argmin reduction.
// One thread per (b, q). Targets staged in LDS. Argmin via packed u64
// (orderedCost<<32 | q) min — deterministic, ties -> smallest q (argmin rule).
// ---------------------------------------------------------------------------
__global__ void main_cost_kernel(const float* __restrict__ boxes,
                                 const float* __restrict__ img,
                                 const float* __restrict__ wsT,
                                 unsigned long long* __restrict__ gmin,
                                 float* __restrict__ Cout) {
  __shared__ float stgt[TT * 16];
  __shared__ float srimg[8];           // reciprocal image sizes
  __shared__ unsigned long long lmin[TT];

  int tid = (int)threadIdx.x;
  int b   = (int)blockIdx.x / 49;
  int q   = ((int)blockIdx.x % 49) * 256 + tid;

  if (tid < (TT * 16) / 4)
    ((float4*)stgt)[tid] = ((const float4*)(wsT + (size_t)b * TT * 16))[tid];
  if (tid < 5)  srimg[tid] = 1.f / img[b * 5 + tid];
  if (tid < TT) lmin[tid] = ~0ull;
  __syncthreads();

  // pred box (layout [bs,5,h,w], fully coalesced)
  float pb[5];
  #pragma unroll
  for (int j = 0; j < 5; ++j) pb[j] = boxes[((size_t)b * 5 + j) * HW + q];
  float th = pb[4] * 0.017453292519943295f;
  float c = cosf(th), s = sinf(th);
  float hw2 = 0.5f * pb[2], hh2 = 0.5f * pb[3];
  float c1x[4], c1y[4];
  c1x[0] = pb[0] + c*hw2 - s*hh2;  c1y[0] = pb[1] + s*hw2 + c*hh2;
  c1x[1] = pb[0] + c*hw2 + s*hh2;  c1y[1] = pb[1] + s*hw2 - c*hh2;
  c1x[2] = pb[0] - c*hw2 + s*hh2;  c1y[2] = pb[1] - s*hw2 - c*hh2;
  c1x[3] = pb[0] - c*hw2 - s*hh2;  c1y[3] = pb[1] - s*hw2 + c*hh2;
  float a1 = pb[2] * pb[3];
  float pbn[5];
  #pragma unroll
  for (int j = 0; j < 5; ++j) pbn[j] = pb[j] * srimg[j];

  // Pull the pre-stored 2*cost_class row (24 contiguous floats) with float4s.
  size_t obase = ((size_t)b * HW + q) * TT;
  float row[TT];
  #pragma unroll
  for (int v = 0; v < TT / 4; ++v)
    *((float4*)&row[v * 4]) = ((const float4*)(Cout + obase))[v];

  #pragma unroll 1
  for (int t = 0; t < TT; ++t) {
    const float* td = &stgt[t * 16];
    float cb = 0.f;
    #pragma unroll
    for (int j = 0; j < 5; ++j) cb += fabsf(pbn[j] - td[9 + j]);
    float inter = quad_inter_area(c1x, c1y, td);
    float iou   = inter * fast_rcp(a1 + td[8] - inter + 1e-8f);
    float cval  = row[t] + 5.f * cb - 2.f * iou;
    row[t] = cval;
    unsigned long long key =
        ((unsigned long long)order_f32(cval) << 32) | (unsigned)q;
    atomicMin(&lmin[t], key);
  }

  // Write the finished C row back (vectorized).
  #pragma unroll
  for (int v = 0; v < TT / 4; ++v)
    ((float4*)(Cout + obase))[v] = *((const float4*)&row[v * 4]);

  __syncthreads();
  if (tid < TT) atomicMin(&gmin[b * TT + tid], lmin[tid]);
}

// ---------------------------------------------------------------------------
// Kernel 3: decode argmin keys -> src_ind / tgt_ind (as floats, per harness)
// ---------------------------------------------------------------------------
__global__ void finalize_kernel(const unsigned long long* __restrict__ gmin,
                                float* __restrict__ out_src,
                                float* __restrict__ out_tgt) {
  int i = (int)threadIdx.x;
  if (i < BS * TT) {
    unsigned q = (unsigned)(gmin[i] & 0xffffffffull);
    out_src[i] = (float)q;
    out_tgt[i] = (float)(i % TT);
  }
}

extern "C" void kernel_launch(void* const* d_in, const int* in_sizes, int n_in,
                              void* d_out, int out_size, void* d_ws, size_t ws_size,
                              hipStream_t stream) {
  const float* logits = (const float*)d_in[0];   // [4,80,112,112]
  const float* boxes  = (const float*)d_in[1];   // [4,5,112,112]
  const int*   labels = (const int*)d_in[2];     // [4,24]
  const float* tboxes = (const float*)d_in[3];   // [4,24,5]
  const float* img    = (const float*)d_in[4];   // [4,5]
  const float* imgt   = (const float*)d_in[5];   // [4,24,5]
  float* out = (float*)d_out;

  unsigned long long* gmin = (unsigned long long*)d_ws;           // 96 u64
  float* wsT = (float*)((char*)d_ws + 1024);                      // 96*16 floats

  const size_t Csize = (size_t)BS * HW * TT;

  prep_kernel<<<1, 128, 0, stream>>>(tboxes, imgt, wsT, gmin);
  class_cost_kernel<<<(BS * QT_PER_B) / 8, 256, 0, stream>>>(logits, labels, out);
  main_cost_kernel<<<BS * 49, 256, 0, stream>>>(boxes, img, wsT, gmin, out);
  finalize_kernel<<<1, 128, 0, stream>>>(gmin, out + Csize, out + Csize + BS * TT);
}